// KGINAggregator_51316269253370
// MI455X (gfx1250) — compile-verified
//
#include <hip/hip_runtime.h>
#include <math.h>

#define N_ENT   100000
#define N_USERS 50000
#define E_KG    1500000
#define NNZ     1000000
#define D       64
#define N_FACT  4
#define N_REL   24

typedef __attribute__((ext_vector_type(2))) float v2f;
typedef __attribute__((ext_vector_type(8))) float v8f;

// ---------------------------------------------------------------------------
// Kernel 1: KG edge scatter. One wave32 per edge; each lane owns 2 of 64 dims.
// msg = rel_h[e] * ent_emb[src], atomically accumulated into neigh_sum[dst].
// ent_emb (25.6 MB) is L2-resident on MI455X (192 MB L2) so gathers are cheap;
// rel_h is a 384 MB stream. Native f32 atomics land in L2.
// ---------------------------------------------------------------------------
__global__ __launch_bounds__(256) void kg_scatter_kernel(
    const float* __restrict__ rel_h, const float* __restrict__ ent_emb,
    const int* __restrict__ kg_src, const int* __restrict__ kg_dst,
    float* __restrict__ neigh_sum, float* __restrict__ deg)
{
    const int edge = blockIdx.x * (blockDim.x >> 5) + (threadIdx.x >> 5);
    const int lane = threadIdx.x & 31;
    if (edge >= E_KG) return;

    const int src = kg_src[edge];
    const int dst = kg_dst[edge];

    const float2 rh = ((const float2*)(rel_h   + (size_t)edge * D))[lane];
    const float2 eh = ((const float2*)(ent_emb + (size_t)src  * D))[lane];

    float* o = neigh_sum + (size_t)dst * D + lane * 2;
    unsafeAtomicAdd(o + 0, rh.x * eh.x);
    unsafeAtomicAdd(o + 1, rh.y * eh.y);
    if (lane == 0) unsafeAtomicAdd(deg + dst, 1.0f);
}

// ---------------------------------------------------------------------------
// Kernel 2: user-item scatter. One wave32 per nnz.
// ---------------------------------------------------------------------------
__global__ __launch_bounds__(256) void ui_scatter_kernel(
    const float* __restrict__ ent_emb, const float* __restrict__ ui_val,
    const int* __restrict__ ui_user, const int* __restrict__ ui_item,
    float* __restrict__ user_agg)
{
    const int nz   = blockIdx.x * (blockDim.x >> 5) + (threadIdx.x >> 5);
    const int lane = threadIdx.x & 31;
    if (nz >= NNZ) return;

    const int u  = ui_user[nz];
    const int it = ui_item[nz];
    const float v = ui_val[nz];

    const float2 e = ((const float2*)(ent_emb + (size_t)it * D))[lane];
    float* o = user_agg + (size_t)u * D + lane * 2;
    unsafeAtomicAdd(o + 0, v * e.x);
    unsafeAtomicAdd(o + 1, v * e.y);
}

// ---------------------------------------------------------------------------
// Kernel 3: normalize neigh = neigh_sum / max(deg, 1)
// ---------------------------------------------------------------------------
__global__ __launch_bounds__(256) void normalize_kernel(
    float* __restrict__ neigh, const float* __restrict__ deg)
{
    const int i = blockIdx.x * blockDim.x + threadIdx.x;
    if (i >= N_ENT * D) return;
    const float d = deg[i >> 6];
    neigh[i] = neigh[i] / fmaxf(d, 1.0f);
}

// ---------------------------------------------------------------------------
// Kernel 4: precompute M = latent_emb^T @ (softmax(disen_att) @ rel_emb).
// mod = (user_emb @ latent^T) @ disen == user_emb @ M  (associativity), so the
// whole modulation collapses to one [U,64]@[64,64] GEMM against this 16 KB
// matrix. Tiny: one block.
// ---------------------------------------------------------------------------
__global__ __launch_bounds__(256) void prep_M_kernel(
    const float* __restrict__ latent_emb, const float* __restrict__ rel_emb,
    const float* __restrict__ disen_att, float* __restrict__ Mmat)
{
    __shared__ float sm[N_FACT][N_REL];
    __shared__ float disen[N_FACT][D];
    const int tid = threadIdx.x;

    if (tid < N_FACT) {
        float mx = -INFINITY;
        for (int r = 0; r < N_REL; ++r) mx = fmaxf(mx, disen_att[tid * N_REL + r]);
        float s = 0.0f;
        for (int r = 0; r < N_REL; ++r) {
            const float e = __expf(disen_att[tid * N_REL + r] - mx);
            sm[tid][r] = e; s += e;
        }
        const float inv = 1.0f / s;
        for (int r = 0; r < N_REL; ++r) sm[tid][r] *= inv;
    }
    __syncthreads();

    {   // disen[f][d] = sum_r sm[f][r] * rel_emb[r][d]   (256 = 4*64 threads)
        const int f = tid >> 6, d = tid & 63;
        float acc = 0.0f;
        for (int r = 0; r < N_REL; ++r) acc += sm[f][r] * rel_emb[r * D + d];
        disen[f][d] = acc;
    }
    __syncthreads();

    // M[i][j] = sum_f latent_emb[f][i] * disen[f][j]
    for (int e = tid; e < D * D; e += 256) {
        const int i = e >> 6, j = e & 63;
        float acc = 0.0f;
        #pragma unroll
        for (int f = 0; f < N_FACT; ++f) acc += latent_emb[f * D + i] * disen[f][j];
        Mmat[e] = acc;
    }
}

// ---------------------------------------------------------------------------
// Kernel 5: WMMA finalize. One wave per 16-user tile (50000 = 3125*16 exact,
// so EXEC is always all-ones as WMMA requires). mod_tile = U_tile @ M via
// V_WMMA_F32_16X16X4_F32: 4 N-tiles x 16 K-steps = 64 wmma ops per wave,
// fused with out = agg * (1 + mod).
//
// ISA layouts (16x16x4 f32):
//  A 16x4 : lanes 0-15 hold rows M=0..15, VGPR0/1 = K0/K1; lanes 16-31 = K2/K3
//  B 4x16 : VGPR0 lanes0-15 = B[0][n], lanes16-31 = B[2][n]; VGPR1 = B[1]/B[3]
//  C/D    : VGPR r, lanes 0-15 -> row r, lanes 16-31 -> row r+8, col = lane%16
// ---------------------------------------------------------------------------
__global__ __launch_bounds__(32) void user_finalize_wmma_kernel(
    const float* __restrict__ user_emb, const float* __restrict__ Mmat,
    float* __restrict__ out_user)
{
    const int tile = blockIdx.x;        // 16 users
    const int lane = threadIdx.x;       // 0..31
    const int half = lane >> 4;         // 0: K even pair, 1: K odd pair
    const int l16  = lane & 15;

    v8f acc[4];
    #pragma unroll
    for (int t = 0; t < 4; ++t) acc[t] = (v8f){0,0,0,0,0,0,0,0};

    const float* U = user_emb + (size_t)tile * 16 * D;

    #pragma unroll
    for (int kb = 0; kb < 16; ++kb) {
        const int k0 = kb * 4 + half * 2;
        v2f a;
        a.x = U[l16 * D + k0 + 0];
        a.y = U[l16 * D + k0 + 1];
        #pragma unroll
        for (int t = 0; t < 4; ++t) {
            v2f b;
            b.x = Mmat[(k0 + 0) * D + t * 16 + l16];
            b.y = Mmat[(k0 + 1) * D + t * 16 + l16];
            acc[t] = __builtin_amdgcn_wmma_f32_16x16x4_f32(
                /*neg_a=*/false, a, /*neg_b=*/false, b,
                /*c_mod=*/(short)0, acc[t], /*reuse_a=*/false, /*reuse_b=*/false);
        }
    }

    // out_user = agg * (1 + mod)
    #pragma unroll
    for (int r = 0; r < 8; ++r) {
        const int row = half ? (r + 8) : r;
        const size_t u = (size_t)tile * 16 + row;
        #pragma unroll
        for (int t = 0; t < 4; ++t) {
            const int col = t * 16 + l16;
            float* p = out_user + u * D + col;
            *p = *p * (1.0f + acc[t][r]);
        }
    }
}

// ---------------------------------------------------------------------------
extern "C" void kernel_launch(void* const* d_in, const int* in_sizes, int n_in,
                              void* d_out, int out_size, void* d_ws, size_t ws_size,
                              hipStream_t stream) {
    const float* ent_emb    = (const float*)d_in[0];
    const float* user_emb   = (const float*)d_in[1];
    const float* latent_emb = (const float*)d_in[2];
    const float* rel_emb    = (const float*)d_in[3];
    const float* disen_att  = (const float*)d_in[4];
    const float* rel_h      = (const float*)d_in[5];
    const float* ui_val     = (const float*)d_in[6];
    const int*   kg_src     = (const int*)d_in[7];
    const int*   kg_dst     = (const int*)d_in[8];
    const int*   ui_user    = (const int*)d_in[9];
    const int*   ui_item    = (const int*)d_in[10];

    float* out_neigh = (float*)d_out;                        // [N_ENT, 64]
    float* out_user  = (float*)d_out + (size_t)N_ENT * D;    // [N_USERS, 64]

    float* deg  = (float*)d_ws;                              // N_ENT floats
    float* Mmat = deg + N_ENT;                               // 64*64 floats

    // Zero accumulators (graph-capture-safe async memsets).
    hipMemsetAsync(d_out, 0, (size_t)(N_ENT + N_USERS) * D * sizeof(float), stream);
    hipMemsetAsync(deg, 0, (size_t)N_ENT * sizeof(float), stream);

    prep_M_kernel<<<1, 256, 0, stream>>>(latent_emb, rel_emb, disen_att, Mmat);

    kg_scatter_kernel<<<(E_KG + 7) / 8, 256, 0, stream>>>(
        rel_h, ent_emb, kg_src, kg_dst, out_neigh, deg);

    ui_scatter_kernel<<<(NNZ + 7) / 8, 256, 0, stream>>>(
        ent_emb, ui_val, ui_user, ui_item, out_user);

    normalize_kernel<<<(N_ENT * D + 255) / 256, 256, 0, stream>>>(out_neigh, deg);

    user_finalize_wmma_kernel<<<N_USERS / 16, 32, 0, stream>>>(
        user_emb, Mmat, out_user);
}